// edge_feature_4982162063990
// MI455X (gfx1250) — compile-verified
//
#include <hip/hip_runtime.h>

// Problem constants (match reference)
#define EF_DIM 256   // feature dim (= GEMM K-reduction = output cols)
#define EF_KNB 32    // neighbors per point (= GEMM M rows)
#define EF_STR 260   // padded LDS row stride for diffT (260 mod 64 = 4 -> conflict-free A frags)

typedef __attribute__((ext_vector_type(2))) float v2f;
typedef __attribute__((ext_vector_type(8))) float v8f;

// One workgroup per (b,n): 256 threads = 8 waves.
// Wave w computes output columns [32w, 32w+32) for all 32 neighbor rows via
// V_WMMA_F32_16X16X4_F32, then reduces max over neighbors.
__global__ __launch_bounds__(256)
void edge_feature_kernel(const float* __restrict__ fi,   // [BN, 256]
                         const float* __restrict__ fj,   // [BN, 256, 32]
                         const float* __restrict__ W1,   // [256, 256]
                         const float* __restrict__ b1,   // [256]
                         const float* __restrict__ W2,   // [256, 256]
                         const float* __restrict__ b2,   // [256]
                         float* __restrict__ out)        // [BN, 256]
{
    __shared__ float fiLds[EF_DIM];
    __shared__ float diffT[EF_KNB * EF_STR];  // diffT[k][d] = fj[d,k] - fi[d]
    __shared__ float vLds[EF_DIM];            // b1+b2 + W1·fi

    const int bn   = blockIdx.x;
    const int t    = threadIdx.x;   // 0..255
    const int lane = t & 31;
    const int wave = t >> 5;        // 0..7

    const float* fi_g = fi + bn * EF_DIM;
    const float* fj_g = fj + bn * (EF_DIM * EF_KNB);

    // ---- Phase A: stage fi and transposed diff tile into LDS ----
    fiLds[t] = fi_g[t];
    #pragma unroll
    for (int i = t; i < EF_DIM * EF_KNB; i += 256) {
        const int d = i >> 5;        // feature index (fj is [d][k], k innermost)
        const int k = i & 31;        // neighbor index
        diffT[k * EF_STR + d] = fj_g[i] - fi_g[d];
    }
    __syncthreads();

    // ---- Phase B: v[t] = b1[t] + b2[t] + sum_d fi[d] * W1[t,d]  (~3% of FLOPs) ----
    {
        float acc = b1[t] + b2[t];
        const float4* wrow = (const float4*)(W1 + t * EF_DIM);
        #pragma unroll 4
        for (int d4 = 0; d4 < EF_DIM / 4; ++d4) {
            const float4 w = wrow[d4];
            const float* f = &fiLds[d4 * 4];
            acc += w.x * f[0] + w.y * f[1] + w.z * f[2] + w.w * f[3];
        }
        vLds[t] = acc;
    }
    __syncthreads();

    // ---- Phase C: C[k, o] = sum_d diffT[k, d] * W2[o, d] via fp32 WMMA ----
    // A-frag (16x4 f32): lanes 0-15 -> K = kd,kd+1 ; lanes 16-31 -> K = kd+2,kd+3
    // B-frag (4x16 f32): same K split across lane halves; N = lane&15
    const int hi    = lane >> 4;       // 0/1: which K pair within the 4-step
    const int lo    = lane & 15;
    const int obase = wave * 32;       // this wave's output-column slab

    v8f acc00 = {}; v8f acc01 = {}; v8f acc10 = {}; v8f acc11 = {};

    const float* w2c0  = W2 + (obase + lo) * EF_DIM;        // col block 0 (o = obase+lo)
    const float* w2c1  = W2 + (obase + 16 + lo) * EF_DIM;   // col block 1
    const float* aRow0 = &diffT[lo * EF_STR];               // neighbor rows 0..15
    const float* aRow1 = &diffT[(lo + 16) * EF_STR];        // neighbor rows 16..31

    #pragma unroll 2
    for (int kd = 0; kd < EF_DIM; kd += 4) {
        const int kk = kd + 2 * hi;
        const v2f a0  = *(const v2f*)(aRow0 + kk);   // LDS, 8B, conflict-free
        const v2f a1  = *(const v2f*)(aRow1 + kk);
        const v2f bm0 = *(const v2f*)(w2c0 + kk);    // global -> L2-resident W2
        const v2f bm1 = *(const v2f*)(w2c1 + kk);
        acc00 = __builtin_amdgcn_wmma_f32_16x16x4_f32(false, a0, false, bm0, (short)0, acc00, false, false);
        acc01 = __builtin_amdgcn_wmma_f32_16x16x4_f32(false, a0, false, bm1, (short)0, acc01, false, false);
        acc10 = __builtin_amdgcn_wmma_f32_16x16x4_f32(false, a1, false, bm0, (short)0, acc10, false, false);
        acc11 = __builtin_amdgcn_wmma_f32_16x16x4_f32(false, a1, false, bm1, (short)0, acc11, false, false);
    }

    // ---- Phase D: out[o] = relu(v[o] + max_k C[k,o])  (relu is monotone) ----
    // C/D layout: VGPR v, lanes 0-15 -> row v (+rb*16), lanes 16-31 -> row v+8; N = lane&15.
    float m0 = acc00[0];
    float m1 = acc01[0];
    #pragma unroll
    for (int v = 1; v < 8; ++v) { m0 = fmaxf(m0, acc00[v]); m1 = fmaxf(m1, acc01[v]); }
    #pragma unroll
    for (int v = 0; v < 8; ++v) { m0 = fmaxf(m0, acc10[v]); m1 = fmaxf(m1, acc11[v]); }
    // merge lane-halves (rows M and M+8 live at same column o in lanes L and L+16)
    m0 = fmaxf(m0, __shfl_xor(m0, 16, 32));
    m1 = fmaxf(m1, __shfl_xor(m1, 16, 32));

    if (lane < 16) {
        const int o0 = obase + lo;
        const int o1 = obase + 16 + lo;
        out[bn * EF_DIM + o0] = fmaxf(vLds[o0] + m0, 0.0f);
        out[bn * EF_DIM + o1] = fmaxf(vLds[o1] + m1, 0.0f);
    }
}

extern "C" void kernel_launch(void* const* d_in, const int* in_sizes, int n_in,
                              void* d_out, int out_size, void* d_ws, size_t ws_size,
                              hipStream_t stream) {
    const float* fi = (const float*)d_in[0];  // f_i_set   [B,N,256]
    const float* fj = (const float*)d_in[1];  // f_j_set_k [B,N,256,32]
    const float* W1 = (const float*)d_in[2];  // [256,256]
    const float* b1 = (const float*)d_in[3];  // [256]
    const float* W2 = (const float*)d_in[4];  // [256,256]
    const float* b2 = (const float*)d_in[5];  // [256]
    float* out = (float*)d_out;               // [B,N,256]

    const int BN = in_sizes[0] / EF_DIM;      // B*N = 20000
    edge_feature_kernel<<<BN, 256, 0, stream>>>(fi, fj, W1, b1, W2, b2, out);
}